// GenerativeModel_39109972197974
// MI455X (gfx1250) — compile-verified
//
#include <hip/hip_runtime.h>

// Problem constants (match reference)
#define K 8
#define D 16
#define NN 32
#define T 200
#define TILE 16
#define NTILES 13                  // ceil(200/16)
#define HALF_LOG2PI 0.91893853320467274178f

typedef __attribute__((ext_vector_type(2))) float v2f;
typedef __attribute__((ext_vector_type(8))) float v8f;

__global__ __launch_bounds__(256) void slds_logprob_kernel(
    const int*   __restrict__ ds,        // (B,T)
    const float* __restrict__ zs,        // (B,T,D)
    const float* __restrict__ obs,       // (B,T,N)
    const float* __restrict__ init_logits,   // (K)
    const float* __restrict__ init_locs,     // (K,D)
    const float* __restrict__ init_lsc,      // (K,D)
    const float* __restrict__ trans_logits,  // (K,K)
    const float* __restrict__ dynA,          // (K,D,D)
    const float* __restrict__ dyn_off,       // (K,D)
    const float* __restrict__ dyn_scale_raw, // (K,D)  used RAW as sigma (per reference)
    const float* __restrict__ emW,           // (K,N,D)
    const float* __restrict__ em_off,        // (K,N)
    const float* __restrict__ em_lsc,        // (K,N)
    float* __restrict__ out)                 // (B)
{
    __shared__ float s_emW[K * NN * D];      // 16 KB
    __shared__ float s_dynA[K * D * D];      // 8 KB
    __shared__ float s_emoff[K * NN];
    __shared__ float s_eminv[K * NN];        // exp(-ls)
    __shared__ float s_emc[K * NN];          // -ls - 0.5*log(2pi)
    __shared__ float s_dynoff[K * D];
    __shared__ float s_dyninv[K * D];        // 1/sigma
    __shared__ float s_dync[K * D];          // -log(sigma) - 0.5*log(2pi)
    __shared__ float s_trans[K * K];
    __shared__ float s_initls[K];
    __shared__ float s_initloc[K * D];
    __shared__ float s_initlsc[K * D];
    __shared__ alignas(16) float s_z[8][17 * D];  // per-wave z rows t0-1 .. t0+15
    __shared__ int   s_s[8][17];             // per-wave states  t0-1 .. t0+15
    __shared__ float s_total;

    const int b    = blockIdx.x;
    const int tid  = threadIdx.x;
    const int wave = tid >> 5;
    const int lane = tid & 31;

    // ---- Stage parameters to LDS, precompute transforms ----
    for (int i = tid; i < K * NN * D; i += 256) s_emW[i] = emW[i];
    for (int i = tid; i < K * D * D; i += 256)  s_dynA[i] = dynA[i];
    for (int i = tid; i < K * NN; i += 256) {
        s_emoff[i] = em_off[i];
        float ls   = em_lsc[i];
        s_eminv[i] = __expf(-ls);
        s_emc[i]   = -ls - HALF_LOG2PI;
    }
    for (int i = tid; i < K * D; i += 256) {
        s_dynoff[i] = dyn_off[i];
        float sc    = dyn_scale_raw[i];          // reference uses this raw as sigma
        s_dyninv[i] = 1.0f / sc;
        s_dync[i]   = -__logf(sc) - HALF_LOG2PI;
        s_initloc[i] = init_locs[i];
        s_initlsc[i] = init_lsc[i];
    }
    if (tid < K) {                               // log_softmax of transition rows
        float m = -1e30f;
        for (int j = 0; j < K; ++j) m = fmaxf(m, trans_logits[tid * K + j]);
        float ssum = 0.f;
        for (int j = 0; j < K; ++j) ssum += __expf(trans_logits[tid * K + j] - m);
        float lse = m + __logf(ssum);
        for (int j = 0; j < K; ++j) s_trans[tid * K + j] = trans_logits[tid * K + j] - lse;
    }
    if (tid == 0) {
        float m = -1e30f;
        for (int j = 0; j < K; ++j) m = fmaxf(m, init_logits[j]);
        float ssum = 0.f;
        for (int j = 0; j < K; ++j) ssum += __expf(init_logits[j] - m);
        float lse = m + __logf(ssum);
        for (int j = 0; j < K; ++j) s_initls[j] = init_logits[j] - lse;
        s_total = 0.f;
    }
    __syncthreads();

    float my_lp = 0.f;
    const int half16 = (lane < 16) ? 0 : 2;      // K-pair offset for A/B fragments
    const int rbase  = (lane < 16) ? 0 : 8;      // C/D row base for this lane half
    const int lcol   = lane & 15;

    // ---- Each wave owns 16-timestep tiles: tt = wave, wave+8, ... ----
    for (int tt = wave; tt < NTILES; tt += 8) {
        const int t0 = tt * TILE;

        if (tt + 8 < NTILES)                     // CDNA5 global_prefetch_b8 hint
            __builtin_prefetch(&zs[((size_t)b * T + (size_t)(tt + 8) * TILE) * D], 0, 1);

        // Stage z rows t0-1 .. t0+15 via async-to-LDS DMA (16B per lane-op).
        float* zb = s_z[wave];
        for (int g = lane; g < 17 * (D / 4); g += 32) {   // 68 float4 groups
            int r  = g >> 2;
            int fo = (g & 3) * 4;
            int t  = max(0, min(T - 1, t0 - 1 + r));
            unsigned ldsoff = (unsigned)(uintptr_t)&zb[r * D + fo];
            unsigned long long ga =
                (unsigned long long)(uintptr_t)&zs[((size_t)b * T + t) * D + fo];
            asm volatile("global_load_async_to_lds_b128 %0, %1, off"
                         :: "v"(ldsoff), "v"(ga) : "memory");
        }
        if (lane < 17) {
            int t = max(0, min(T - 1, t0 - 1 + lane));
            s_s[wave][lane] = ds[(size_t)b * T + t];
        }
        asm volatile("s_wait_asynccnt 0" ::: "memory");
        asm volatile("s_wait_dscnt 0" ::: "memory");

        // A-fragments, f32 16x4 ISA layout: lanes 0-15 hold K=4c,4c+1; lanes 16-31 K=4c+2,4c+3
        v2f a_em[4], a_dyn[4];
        #pragma unroll
        for (int c = 0; c < 4; ++c) {
            int dbase = 4 * c + half16;
            a_em[c].x  = zb[(lcol + 1) * D + dbase];      // z[t0+row]
            a_em[c].y  = zb[(lcol + 1) * D + dbase + 1];
            a_dyn[c].x = zb[lcol * D + dbase];            // z[t0+row-1]
            a_dyn[c].y = zb[lcol * D + dbase + 1];
        }

        // ---- Per-tile, per-lane invariants (hoisted out of k-loop) ----
        float obs0[8], obs1[8], ztgt[8], vem[8], vdy[8];
        int   st[8];
        #pragma unroll
        for (int j = 0; j < 8; ++j) {
            int row = j + rbase;
            int t   = t0 + row;
            int tc  = min(t, T - 1);
            st[j]   = s_s[wave][row + 1];
            ztgt[j] = zb[(row + 1) * D + lcol];
            obs0[j] = obs[((size_t)b * T + tc) * NN + lcol];
            obs1[j] = obs[((size_t)b * T + tc) * NN + lcol + 16];
            vem[j]  = (t < T) ? 1.f : 0.f;
            vdy[j]  = (t >= 1 && t < T) ? 1.f : 0.f;
        }

        // ---- Compute against ALL K matrices; branch-free masked accumulate ----
        for (int k = 0; k < K; ++k) {
            // per-(k,lane) parameters, loaded once per k
            float dyoff = s_dynoff[k * D + lcol];
            float dyinv = s_dyninv[k * D + lcol];
            float dyc   = s_dync  [k * D + lcol];
            float eoff0 = s_emoff[k * NN + lcol];
            float einv0 = s_eminv[k * NN + lcol];
            float ec0   = s_emc  [k * NN + lcol];
            float eoff1 = s_emoff[k * NN + lcol + 16];
            float einv1 = s_eminv[k * NN + lcol + 16];
            float ec1   = s_emc  [k * NN + lcol + 16];

            // dynamics: dyn_loc[row,i] = sum_d z_prev[row,d] * A[k][i][d]
            v8f accd = {0.f, 0.f, 0.f, 0.f, 0.f, 0.f, 0.f, 0.f};
            #pragma unroll
            for (int c = 0; c < 4; ++c) {
                int dbase = 4 * c + half16;
                v2f bf;
                bf.x = s_dynA[(k * D + lcol) * D + dbase];
                bf.y = s_dynA[(k * D + lcol) * D + dbase + 1];
                accd = __builtin_amdgcn_wmma_f32_16x16x4_f32(
                    false, a_dyn[c], false, bf, (short)0, accd, false, false);
            }
            // emission n-tile 0 and 1
            v8f acce0 = {0.f, 0.f, 0.f, 0.f, 0.f, 0.f, 0.f, 0.f};
            v8f acce1 = {0.f, 0.f, 0.f, 0.f, 0.f, 0.f, 0.f, 0.f};
            #pragma unroll
            for (int c = 0; c < 4; ++c) {
                int dbase = 4 * c + half16;
                v2f bf0, bf1;
                bf0.x = s_emW[(k * NN + lcol) * D + dbase];
                bf0.y = s_emW[(k * NN + lcol) * D + dbase + 1];
                bf1.x = s_emW[(k * NN + lcol + 16) * D + dbase];
                bf1.y = s_emW[(k * NN + lcol + 16) * D + dbase + 1];
                acce0 = __builtin_amdgcn_wmma_f32_16x16x4_f32(
                    false, a_em[c], false, bf0, (short)0, acce0, false, false);
                acce1 = __builtin_amdgcn_wmma_f32_16x16x4_f32(
                    false, a_em[c], false, bf1, (short)0, acce1, false, false);
            }

            #pragma unroll
            for (int j = 0; j < 8; ++j) {
                float mdy = (st[j] == k) ? vdy[j] : 0.f;   // v_cmp + v_cndmask
                float mem = (st[j] == k) ? vem[j] : 0.f;
                // dynamics term
                float rd  = (ztgt[j] - (accd[j] + dyoff)) * dyinv;
                my_lp = fmaf(mdy, fmaf(-0.5f * rd, rd, dyc), my_lp);
                // emission terms (both 16-col tiles)
                float r0  = (obs0[j] - (acce0[j] + eoff0)) * einv0;
                my_lp = fmaf(mem, fmaf(-0.5f * r0, r0, ec0), my_lp);
                float r1  = (obs1[j] - (acce1[j] + eoff1)) * einv1;
                my_lp = fmaf(mem, fmaf(-0.5f * r1, r1, ec1), my_lp);
            }
        }

        // ---- transition terms: one per valid t in this tile (lanes 0..15)
        if (lane < 16) {
            int t = t0 + lane;
            if (t >= 1 && t < T)
                my_lp += s_trans[s_s[wave][lane] * K + s_s[wave][lane + 1]];
        }
        // ---- init terms (tile 0 only)
        if (tt == 0 && lane < D) {
            int   s0  = s_s[wave][1];
            float lsc = s_initlsc[s0 * D + lane];
            float r   = (zb[1 * D + lane] - s_initloc[s0 * D + lane]) * __expf(-lsc);
            my_lp += -0.5f * r * r - lsc - HALF_LOG2PI;
            if (lane == 0) my_lp += s_initls[s0];
        }
    }

    // ---- wave32 reduction, then block accumulate via ds_add_f32 ----
    #pragma unroll
    for (int off = 16; off > 0; off >>= 1)
        my_lp += __shfl_xor(my_lp, off, 32);
    if (lane == 0) atomicAdd(&s_total, my_lp);
    __syncthreads();
    if (tid == 0) out[b] = s_total;
}

extern "C" void kernel_launch(void* const* d_in, const int* in_sizes, int n_in,
                              void* d_out, int out_size, void* d_ws, size_t ws_size,
                              hipStream_t stream) {
    const int*   dsp         = (const int*)  d_in[0];
    const float* zs          = (const float*)d_in[1];
    const float* obsp        = (const float*)d_in[2];
    const float* init_logits = (const float*)d_in[3];
    const float* init_locs   = (const float*)d_in[4];
    const float* init_lsc    = (const float*)d_in[5];
    const float* trans       = (const float*)d_in[6];
    const float* dynA        = (const float*)d_in[7];
    const float* dyn_off     = (const float*)d_in[8];
    const float* dyn_scale   = (const float*)d_in[9];
    const float* emW         = (const float*)d_in[10];
    const float* em_off      = (const float*)d_in[11];
    const float* em_lsc      = (const float*)d_in[12];
    float*       outp        = (float*)d_out;

    const int B = in_sizes[0] / T;   // (B,T) discrete_states
    slds_logprob_kernel<<<B, 256, 0, stream>>>(
        dsp, zs, obsp, init_logits, init_locs, init_lsc, trans,
        dynA, dyn_off, dyn_scale, emW, em_off, em_lsc, outp);
}